// DepthWise_81432579932540
// MI455X (gfx1250) — compile-verified
//
#include <hip/hip_runtime.h>

typedef __attribute__((ext_vector_type(2))) float v2f;
typedef __attribute__((ext_vector_type(8))) float v8f;

#define HH 512
#define WW 512
#define OHH 510
#define OWW 510
#define NC 16
#define NB 4
#define NKP 9

__global__ __launch_bounds__(256) void dw9_wmma_kernel(
    const float* __restrict__ x,    // [4,16,512,512] one branch's input
    const float* __restrict__ w,    // [9,16,3,3]
    const float* __restrict__ mk,   // [9,16,3,3]
    float* __restrict__ out)        // [9,4,16,510,510] (branch offset applied by caller)
{
    // block -> (b, c, y); blockIdx.x = (b*16 + c)*510 + y
    const int bid = blockIdx.x;
    const int y   = bid % OHH;
    const int bc  = bid / OHH;
    const int c   = bc % NC;
    const int b   = bc / NC;

    const int tid  = threadIdx.x;
    const int wave = tid >> 5;   // 0..7
    const int lane = tid & 31;
    const int n    = lane & 15;  // N (pixel column) / M (pattern row) index
    const int half = lane >> 4;  // K split: lanes 0-15 hold K={0,1}, lanes 16-31 K={2,3}

    // ---- A matrix: (w*m)[pattern = n][tap = K], zero-padded (16 rows x 12 K) ----
    v2f A0, A1, A2;
    {
        const int kb0 = half * 2;       // taps {0,1} or {2,3}
        const int kb1 = 4 + half * 2;   // taps {4,5} or {6,7}
        const int wrow = (n < NKP) ? (n * NC + c) * 9 : 0;
        const bool mrow = (n < NKP);
        A0[0] = mrow ? w[wrow + kb0 + 0] * mk[wrow + kb0 + 0] : 0.0f;
        A0[1] = mrow ? w[wrow + kb0 + 1] * mk[wrow + kb0 + 1] : 0.0f;
        A1[0] = mrow ? w[wrow + kb1 + 0] * mk[wrow + kb1 + 0] : 0.0f;
        A1[1] = mrow ? w[wrow + kb1 + 1] * mk[wrow + kb1 + 1] : 0.0f;
        A2[0] = (mrow && !half) ? w[wrow + 8] * mk[wrow + 8] : 0.0f; // only K=8 real
        A2[1] = 0.0f;
    }

    // Uniform (SGPR) channel base; per-lane 32-bit element indices -> saddr+voffset form.
    const float* __restrict__ xch = x + ((size_t)(b * NC + c)) * HH * WW;
    const int yW = y * WW;

    // Per-lane static tap offsets (half-dependent), hoisted out of the tile loop.
    //   B0.v0: tap(0,0)|(0,2)   B0.v1: (0,1)|(1,0)
    //   B1.v0: (1,1)|(2,0)      B1.v1: (1,2)|(2,1)
    //   B2.v0: (2,2)|pad        B2.v1: pad
    const int o00 = half ? 2          : 0;
    const int o01 = half ? WW         : 1;
    const int o10 = half ? 2 * WW     : WW + 1;
    const int o11 = half ? 2 * WW + 1 : WW + 2;
    const int o20 = 2 * WW + 2;

    const int plane  = OHH * OWW;          // 260100
    const int kstepI = NB * NC * plane;    // 16,646,400; 9*kstepI < 2^31
    const int obaseI = (b * NC + c) * plane + y * OWW;

    // 32 column tiles of 16 output pixels; 8 waves stride across them
    for (int tx = wave; tx < 32; tx += 8) {
        const int x0    = tx * 16;
        const int col   = x0 + n;
        const bool valid = (col < OWW);
        const int colL  = valid ? col : (OWW - 1);   // clamp: loads always in-bounds
        const int base  = yW + colL;

        // ---- B matrix: 5 unconditional loads, zero-pad taps 9..11 ----
        v2f B0, B1, B2;
        B0[0] = xch[base + o00];
        B0[1] = xch[base + o01];
        B1[0] = xch[base + o10];
        B1[1] = xch[base + o11];
        const float b20 = xch[base + o20];
        B2[0] = half ? 0.0f : b20;   // one v_cndmask, no exec branch
        B2[1] = 0.0f;

        // ---- D = A(16x12) * B(12x16), exact fp32, via 3 chained WMMAs ----
        v8f acc = {};
        acc = __builtin_amdgcn_wmma_f32_16x16x4_f32(false, A0, false, B0, (short)0, acc, false, false);
        acc = __builtin_amdgcn_wmma_f32_16x16x4_f32(false, A1, false, B1, (short)0, acc, false, false);
        acc = __builtin_amdgcn_wmma_f32_16x16x4_f32(false, A2, false, B2, (short)0, acc, false, false);

        // ---- clamp(-128,127) + round-to-nearest-even, nontemporal stream out ----
        // D layout: VGPR r -> pattern r (lanes 0-15), pattern 8+r (lanes 16-31).
        if (valid) {
            const int oidx = obaseI + col;
            if (half == 0) {
#pragma unroll
                for (int r = 0; r < 8; ++r) {
                    const float v0 = __builtin_amdgcn_fmed3f(acc[r], -128.0f, 127.0f);
                    __builtin_nontemporal_store(rintf(v0), out + (oidx + r * kstepI));
                }
            } else {
                // pattern 8 = VGPR0 of lanes 16-31
                const float v8 = __builtin_amdgcn_fmed3f(acc[0], -128.0f, 127.0f);
                __builtin_nontemporal_store(rintf(v8), out + (oidx + 8 * kstepI));
            }
        }
    }
}

extern "C" void kernel_launch(void* const* d_in, const int* in_sizes, int n_in,
                              void* d_out, int out_size, void* d_ws, size_t ws_size,
                              hipStream_t stream) {
    const float* xh = (const float*)d_in[0];
    const float* xl = (const float*)d_in[1];
    const float* wh = (const float*)d_in[2];
    const float* wl = (const float*)d_in[3];
    const float* mh = (const float*)d_in[4];
    const float* ml = (const float*)d_in[5];
    float* out = (float*)d_out;

    const size_t branch_stride = (size_t)NKP * NB * NC * OHH * OWW;
    dim3 grid(NB * NC * OHH), block(256);
    dw9_wmma_kernel<<<grid, block, 0, stream>>>(xh, wh, mh, out);
    dw9_wmma_kernel<<<grid, block, 0, stream>>>(xl, wl, ml, out + branch_stride);
}